// GraphConstructor_71519795413811
// MI455X (gfx1250) — compile-verified
//
#include <hip/hip_runtime.h>
#include <hip/hip_bf16.h>
#include <math.h>

// ---------------------------------------------------------------------------
// Problem constants (fixed by the reference)
// ---------------------------------------------------------------------------
#define BATCH   1024
#define K_IN    81920     // IN_FEAT
#define N_OUT   320       // C*T
#define C_CH    32
#define T_N     10
#define CB      16

// GEMM tiling
#define SPLITK  8
#define KCHUNK  (K_IN / SPLITK)   // 10240
#define KT      16                // K staged per LDS refill
#define NTILES  (KCHUNK / KT)     // 640
#define MT      128               // rows per block (8 waves x 16)
#define NT      64                // cols per block (4 wmma tiles per wave)
#define LDS_PAD 20                // padded row stride (floats): 80B -> 16B-aligned, bank-conflict free

typedef __attribute__((ext_vector_type(2))) float v2f;
typedef __attribute__((ext_vector_type(8))) float v8f;

// ---------------------------------------------------------------------------
// CDNA5 async-to-LDS helpers (inline asm; portable across both toolchains).
// GLOBAL_LOAD_ASYNC_TO_LDS_B128: LDS byte offset in a VGPR, 64-bit global
// address in a VGPR pair, tracked with ASYNCcnt (CDNA5 ISA).
// ---------------------------------------------------------------------------
__device__ __forceinline__ void async_copy16(unsigned int ldsByte, const float* g)
{
    asm volatile("global_load_async_to_lds_b128 %0, %1, off"
                 :: "v"(ldsByte), "v"(g)
                 : "memory");
}
__device__ __forceinline__ void wait_async0()
{
    asm volatile("s_wait_asynccnt 0x0" ::: "memory");
}
// Low 32 bits of a flat LDS pointer == LDS byte offset (flat aperture rule).
__device__ __forceinline__ unsigned int lds_off(const void* p)
{
    return (unsigned int)(size_t)p;
}

// ---------------------------------------------------------------------------
// K1: split-K GEMM  pp[s][m][n] = sum_{k in chunk s} x[m,k] * W_lin[n,k]
// V_WMMA_F32_16X16X4_F32, double-buffered LDS fed by async-to-LDS copies.
// ---------------------------------------------------------------------------
__global__ __launch_bounds__(256) void gemm_splitk_wmma(
    const float* __restrict__ x,      // [1024, 81920]
    const float* __restrict__ W,      // [320, 81920]  (B[k,n] = W[n,k])
    float* __restrict__ pp)           // [SPLITK, 1024, 320]
{
    __shared__ float As[2][MT * LDS_PAD];   // A tiles [128][16] padded
    __shared__ float Bs[2][NT * LDS_PAD];   // B tiles [64][16]  padded

    const int tid  = threadIdx.x;
    const int lane = tid & 31;
    const int wv   = tid >> 5;          // 0..7 : wave's 16-row band
    const int l16  = lane & 15;
    const int hi   = lane >> 4;         // 0: K+0/1, 1: K+2/3 (ISA A/B layout)

    const int m0 = (blockIdx.x & 7) * MT;        // 8 m-blocks
    const int n0 = (blockIdx.x >> 3) * NT;       // 5 n-blocks
    const int s  = blockIdx.y;                   // split-K index
    const int kbase0 = s * KCHUNK;

    // ---- per-thread staging assignments (fixed across iterations) ----
    // A tile: 128 rows x 16 k = 512 float4 chunks, 2 per thread.
    const int aRow0 = tid >> 2;                  // chunks c = tid
    const int aKq0  = (tid & 3) * 4;
    const int aRow1 = (tid + 256) >> 2;          // chunks c = tid + 256
    const int aKq1  = ((tid + 256) & 3) * 4;
    // B tile: 64 rows x 16 k = 256 float4 chunks, 1 per thread.
    const int bRow  = tid >> 2;
    const int bKq   = (tid & 3) * 4;

    const float* gA0 = x + (size_t)(m0 + aRow0) * K_IN + kbase0 + aKq0;
    const float* gA1 = x + (size_t)(m0 + aRow1) * K_IN + kbase0 + aKq1;
    const float* gB  = W + (size_t)(n0 + bRow ) * K_IN + kbase0 + bKq;

    const unsigned int lA0[2] = { lds_off(&As[0][aRow0 * LDS_PAD + aKq0]),
                                  lds_off(&As[1][aRow0 * LDS_PAD + aKq0]) };
    const unsigned int lA1[2] = { lds_off(&As[0][aRow1 * LDS_PAD + aKq1]),
                                  lds_off(&As[1][aRow1 * LDS_PAD + aKq1]) };
    const unsigned int lB [2] = { lds_off(&Bs[0][bRow  * LDS_PAD + bKq ]),
                                  lds_off(&Bs[1][bRow  * LDS_PAD + bKq ]) };

    v8f acc0 = {}; v8f acc1 = {}; v8f acc2 = {}; v8f acc3 = {};

    const int aRowOff = (wv * 16 + l16) * LDS_PAD;

    // ---- prologue: stage tile 0 into buffer 0 ----
    async_copy16(lA0[0], gA0);
    async_copy16(lA1[0], gA1);
    async_copy16(lB [0], gB );

    for (int i = 0; i < NTILES; ++i) {
        const int cur = i & 1;
        const int nxt = cur ^ 1;

        // tile i landed (own asyncs), publish to all waves
        wait_async0();
        __syncthreads();

        // overlap: stage tile i+1 into the other buffer while computing tile i
        if (i + 1 < NTILES) {
            const int koff = (i + 1) * KT;
            async_copy16(lA0[nxt], gA0 + koff);
            async_copy16(lA1[nxt], gA1 + koff);
            async_copy16(lB [nxt], gB  + koff);
        }

        // ---- 4 WMMA k-steps x 4 n-tiles on buffer `cur` ----
        const float* __restrict__ Ab = &As[cur][0];
        const float* __restrict__ Bb = &Bs[cur][0];
        #pragma unroll
        for (int kk = 0; kk < KT; kk += 4) {
            const int ko = kk + hi * 2;
            const v2f a = *(const v2f*)(&Ab[aRowOff + ko]);

            v2f b0 = *(const v2f*)(&Bb[( 0 + l16) * LDS_PAD + ko]);
            acc0 = __builtin_amdgcn_wmma_f32_16x16x4_f32(false, a, false, b0, (short)0, acc0, false, false);
            v2f b1 = *(const v2f*)(&Bb[(16 + l16) * LDS_PAD + ko]);
            acc1 = __builtin_amdgcn_wmma_f32_16x16x4_f32(false, a, false, b1, (short)0, acc1, false, false);
            v2f b2 = *(const v2f*)(&Bb[(32 + l16) * LDS_PAD + ko]);
            acc2 = __builtin_amdgcn_wmma_f32_16x16x4_f32(false, a, false, b2, (short)0, acc2, false, false);
            v2f b3 = *(const v2f*)(&Bb[(48 + l16) * LDS_PAD + ko]);
            acc3 = __builtin_amdgcn_wmma_f32_16x16x4_f32(false, a, false, b3, (short)0, acc3, false, false);
        }

        // all waves done reading `cur` before it is restaged next iteration
        __syncthreads();
    }

    // ---- write partials. C layout: VGPR r -> M = base + r (+8 for upper half),
    //      lane%16 -> N. ----
    const int mBase = m0 + wv * 16 + hi * 8;
    float* outS = pp + (size_t)s * BATCH * N_OUT;
    #pragma unroll
    for (int r = 0; r < 8; ++r) {
        float* rowp = outS + (size_t)(mBase + r) * N_OUT + n0;
        rowp[ 0 + l16] = acc0[r];
        rowp[16 + l16] = acc1[r];
        rowp[32 + l16] = acc2[r];
        rowp[48 + l16] = acc3[r];
    }
}

// ---------------------------------------------------------------------------
// K2: reduce split-K partials + bias:  feat[m,n] = b_lin[n] + sum_s pp[s][m][n]
// ---------------------------------------------------------------------------
__global__ void reduce_bias(const float* __restrict__ pp,
                            const float* __restrict__ b_lin,
                            float* __restrict__ feat)
{
    const int idx = blockIdx.x * blockDim.x + threadIdx.x;
    if (idx >= BATCH * N_OUT) return;
    float acc = b_lin[idx % N_OUT];
    #pragma unroll
    for (int s = 0; s < SPLITK; ++s)
        acc += pp[(size_t)s * BATCH * N_OUT + idx];
    feat[idx] = acc;
}

// ---------------------------------------------------------------------------
// K3: branch convs + attention scores.  One block per batch element.
// ---------------------------------------------------------------------------
__global__ __launch_bounds__(160) void branch_scores(
    const float* __restrict__ feat,
    const float* __restrict__ Wb1, const float* __restrict__ bb1,
    const float* __restrict__ Wb2, const float* __restrict__ bb2,
    float* __restrict__ scores)
{
    __shared__ float Xs[N_OUT];
    __shared__ float x1s[CB * T_N];
    __shared__ float x2s[CB * T_N];

    const int b   = blockIdx.x;
    const int tid = threadIdx.x;              // 0..159

    Xs[tid]       = feat[(size_t)b * N_OUT + tid];
    Xs[tid + 160] = feat[(size_t)b * N_OUT + tid + 160];
    __syncthreads();

    const int o = tid / T_N;                  // 0..15
    const int t = tid % T_N;
    float s1 = bb1[o], s2 = bb2[o];
    #pragma unroll
    for (int c = 0; c < C_CH; ++c) {
        const float xv = Xs[c * T_N + t];
        s1 = fmaf(Wb1[o * C_CH + c], xv, s1);
        s2 = fmaf(Wb2[o * C_CH + c], xv, s2);
    }
    x1s[tid] = fmaxf(s1, 0.0f);
    x2s[tid] = fmaxf(s2, 0.0f);
    __syncthreads();

    if (tid < T_N * T_N) {
        const int tt = tid / T_N, ss = tid % T_N;
        float acc = 0.0f;
        #pragma unroll
        for (int c = 0; c < CB; ++c)
            acc = fmaf(x1s[c * T_N + tt], x2s[c * T_N + ss], acc);
        scores[(size_t)b * 100 + tid] = acc;
    }
}

// ---------------------------------------------------------------------------
// K4: softmax over the BATCH axis per (t,s), then add exp(-|t-s|).
// ---------------------------------------------------------------------------
__global__ __launch_bounds__(256) void softmax_batch(
    const float* __restrict__ scores, float* __restrict__ A)
{
    __shared__ float red[256];
    const int p   = blockIdx.x;               // 0..99
    const int tid = threadIdx.x;

    float m = -INFINITY;
    for (int b = tid; b < BATCH; b += 256)
        m = fmaxf(m, scores[(size_t)b * 100 + p]);
    red[tid] = m; __syncthreads();
    for (int st = 128; st > 0; st >>= 1) {
        if (tid < st) red[tid] = fmaxf(red[tid], red[tid + st]);
        __syncthreads();
    }
    const float mx = red[0]; __syncthreads();

    float sum = 0.0f;
    for (int b = tid; b < BATCH; b += 256)
        sum += expf(scores[(size_t)b * 100 + p] - mx);
    red[tid] = sum; __syncthreads();
    for (int st = 128; st > 0; st >>= 1) {
        if (tid < st) red[tid] += red[tid + st];
        __syncthreads();
    }
    const float inv = 1.0f / red[0];

    const int t = p / T_N, s = p % T_N;
    const float a2 = expf(-fabsf((float)(t - s)));
    for (int b = tid; b < BATCH; b += 256)
        A[(size_t)b * 100 + p] = expf(scores[(size_t)b * 100 + p] - mx) * inv + a2;
}

// ---------------------------------------------------------------------------
// K5: 3x graph conv (residual) + classifier + sigmoid. One block per batch.
// ---------------------------------------------------------------------------
__global__ __launch_bounds__(320) void graph_cls(
    const float* __restrict__ feat, const float* __restrict__ Amat,
    const float* __restrict__ Wg1, const float* __restrict__ bg1,
    const float* __restrict__ Wg2, const float* __restrict__ bg2,
    const float* __restrict__ Wg3, const float* __restrict__ bg3,
    const float* __restrict__ Wcls, const float* __restrict__ bcls,
    float* __restrict__ out)
{
    __shared__ float Xc[N_OUT];
    __shared__ float Xl[N_OUT];
    __shared__ float Ash[100];
    __shared__ float red[N_OUT];

    const int b   = blockIdx.x;
    const int tid = threadIdx.x;              // 0..319
    const int c   = tid / T_N;
    const int t   = tid % T_N;

    Xc[tid] = feat[(size_t)b * N_OUT + tid];
    if (tid < 100) Ash[tid] = Amat[(size_t)b * 100 + tid];
    __syncthreads();

    const float* Wg[3] = {Wg1, Wg2, Wg3};
    const float* bg[3] = {bg1, bg2, bg3};

    #pragma unroll
    for (int l = 0; l < 3; ++l) {
        float acc = 0.0f;
        #pragma unroll
        for (int u = 0; u < T_N; ++u)
            acc = fmaf(Xc[c * T_N + u], Ash[u * T_N + t], acc);
        Xl[tid] = acc;
        __syncthreads();
        float o = bg[l][c];
        #pragma unroll
        for (int u = 0; u < C_CH; ++u)
            o = fmaf(Wg[l][c * C_CH + u], Xl[u * T_N + t], o);
        o += Xc[tid];
        __syncthreads();
        Xc[tid] = o;
        __syncthreads();
    }

    red[tid] = Xc[tid] * Wcls[tid];
    __syncthreads();
    if (tid < 64) red[tid] += red[tid + 256];
    __syncthreads();
    for (int st = 128; st > 0; st >>= 1) {
        if (tid < st) red[tid] += red[tid + st];
        __syncthreads();
    }
    if (tid == 0) {
        const float lg = red[0] + bcls[0];
        out[b] = lg;                                     // logits
        out[BATCH + b] = 1.0f / (1.0f + expf(-lg));      // sigmoid
    }
}

// ---------------------------------------------------------------------------
// Host launcher
// ---------------------------------------------------------------------------
extern "C" void kernel_launch(void* const* d_in, const int* in_sizes, int n_in,
                              void* d_out, int out_size, void* d_ws, size_t ws_size,
                              hipStream_t stream)
{
    const float* x     = (const float*)d_in[0];
    const float* W_lin = (const float*)d_in[1];
    const float* b_lin = (const float*)d_in[2];
    const float* W_b1  = (const float*)d_in[3];
    const float* b_b1  = (const float*)d_in[4];
    const float* W_b2  = (const float*)d_in[5];
    const float* b_b2  = (const float*)d_in[6];
    const float* W_g1  = (const float*)d_in[7];
    const float* b_g1  = (const float*)d_in[8];
    const float* W_g2  = (const float*)d_in[9];
    const float* b_g2  = (const float*)d_in[10];
    const float* W_g3  = (const float*)d_in[11];
    const float* b_g3  = (const float*)d_in[12];
    const float* W_cls = (const float*)d_in[13];
    const float* b_cls = (const float*)d_in[14];

    float* ws     = (float*)d_ws;
    float* pp     = ws;                                  // SPLITK*1024*320
    float* feat   = pp + (size_t)SPLITK * BATCH * N_OUT; // 1024*320
    float* scores = feat + (size_t)BATCH * N_OUT;        // 1024*100
    float* Amat   = scores + (size_t)BATCH * 100;        // 1024*100

    dim3 g1(40, SPLITK);
    gemm_splitk_wmma<<<g1, 256, 0, stream>>>(x, W_lin, pp);

    reduce_bias<<<(BATCH * N_OUT + 255) / 256, 256, 0, stream>>>(pp, b_lin, feat);

    branch_scores<<<BATCH, 160, 0, stream>>>(feat, W_b1, b_b1, W_b2, b_b2, scores);

    softmax_batch<<<100, 256, 0, stream>>>(scores, Amat);

    graph_cls<<<BATCH, 320, 0, stream>>>(feat, Amat, W_g1, b_g1, W_g2, b_g2,
                                         W_g3, b_g3, W_cls, b_cls, (float*)d_out);
}